// MessagePassing_84920093376843
// MI455X (gfx1250) — compile-verified
//
#include <hip/hip_runtime.h>
#include <hip/hip_bf16.h>

typedef __attribute__((ext_vector_type(2))) float v2f;
typedef __attribute__((ext_vector_type(8))) float v8f;

#define NB 4
#define NA 2048
#define NK 64
#define NF 256
#define NR 128
#define LOG2F32 0.6931471805599453f

__device__ __forceinline__ float sspf(float x) {
    // softplus(x) - log(2), numerically stable
    float ax = fabsf(x);
    return fmaxf(x, 0.0f) + logf(1.0f + expf(-ax)) - LOG2F32;
}

__device__ __forceinline__ float poly_cutoff(float d) {
    float x = d * (1.0f / 5.0f);            // CUTOFF = 5
    float x2 = x * x;
    float x4 = x2 * x2;
    float x9 = x4 * x4 * x;                 // x^9
    float f = 1.0f - 55.0f * x9 + 99.0f * x9 * x - 45.0f * x9 * x2;
    return (x < 1.0f) ? f : 0.0f;
}

// ---------------------------------------------------------------------------
// Kernel 1: a_msij = ssp(a @ W1 + b1) @ W2 + b2      (8192 x 256)
// One block per 16-row strip. 8 waves; wave w computes n-tiles 2w, 2w+1.
// fp32 WMMA (V_WMMA_F32_16X16X4_F32), A-frags from padded LDS.
// ---------------------------------------------------------------------------
#define PITCH1 260   // 256 + 4 pad -> conflict-free column reads (260 % 64 == 4)

__global__ __launch_bounds__(256) void k_amsij(
    const float* __restrict__ a,
    const float* __restrict__ W1, const float* __restrict__ b1,
    const float* __restrict__ W2, const float* __restrict__ b2,
    float* __restrict__ amsij)
{
    __shared__ float sA[16 * PITCH1];
    __shared__ float sH[16 * PITCH1];

    const int tid  = threadIdx.x;
    const int row0 = blockIdx.x * 16;

    // cooperative load of 16x256 input strip
    #pragma unroll
    for (int i = 0; i < 16; ++i)
        sA[i * PITCH1 + tid] = a[(row0 + i) * NF + tid];
    __syncthreads();

    const int wave = tid >> 5;
    const int lane = tid & 31;
    const int l16  = lane & 15;
    const int hi   = lane >> 4;

    // ---- phase 1: h = ssp(A @ W1 + b1) ----
    for (int t = 0; t < 2; ++t) {
        const int nt   = wave * 2 + t;
        const int fcol = nt * 16 + l16;
        v8f acc = {};
        for (int kc = 0; kc < NF / 4; ++kc) {
            const int k0 = kc * 4 + 2 * hi;
            v2f af = *(const v2f*)(&sA[l16 * PITCH1 + k0]);
            v2f bf;
            bf.x = W1[(k0    ) * NF + fcol];
            bf.y = W1[(k0 + 1) * NF + fcol];
            acc = __builtin_amdgcn_wmma_f32_16x16x4_f32(
                false, af, false, bf, (short)0, acc, false, false);
        }
        const float bias = b1[fcol];
        #pragma unroll
        for (int v = 0; v < 8; ++v)
            sH[(v + 8 * hi) * PITCH1 + fcol] = sspf(acc[v] + bias);
    }
    __syncthreads();

    // ---- phase 2: a_msij = H @ W2 + b2 ----
    for (int t = 0; t < 2; ++t) {
        const int nt   = wave * 2 + t;
        const int fcol = nt * 16 + l16;
        v8f acc = {};
        for (int kc = 0; kc < NF / 4; ++kc) {
            const int k0 = kc * 4 + 2 * hi;
            v2f af = *(const v2f*)(&sH[l16 * PITCH1 + k0]);
            v2f bf;
            bf.x = W2[(k0    ) * NF + fcol];
            bf.y = W2[(k0 + 1) * NF + fcol];
            acc = __builtin_amdgcn_wmma_f32_16x16x4_f32(
                false, af, false, bf, (short)0, acc, false, false);
        }
        const float bias = b2[fcol];
        #pragma unroll
        for (int v = 0; v < 8; ++v)
            amsij[(row0 + v + 8 * hi) * NF + fcol] = acc[v] + bias;
    }
}

// ---------------------------------------------------------------------------
// Kernel 2: fused  rbf@W_rbf -> *cutoff*NM*ai*aj -> p_out, a_out
// One block per (b, a) row. msij tile is 64 x 256; wave w owns 2 n-tiles,
// all 4 m-tiles (B fragments reused across m-tiles).
// Streaming traffic (rbf, p, p_out) uses non-temporal hints so the 192 MB L2
// stays reserved for the 8 MB a_msij gather table + weight matrices.
// ---------------------------------------------------------------------------
#define PITCH2 132   // 128 + 4 pad (132 % 64 == 4) -> conflict-free A-frag reads

__global__ __launch_bounds__(256) void k_msg(
    const float* __restrict__ a,
    const float* __restrict__ p,
    const float* __restrict__ rbf,
    const float* __restrict__ D,
    const int*   __restrict__ N,
    const float* __restrict__ NM,
    const float* __restrict__ Wr,
    const float* __restrict__ br,
    const float* __restrict__ amsij,
    float* __restrict__ a_out,
    float* __restrict__ p_out)
{
    __shared__ float sR[NK * PITCH2];
    __shared__ float cd[NK];
    __shared__ int   nidx[NK];

    const int tid = threadIdx.x;
    const int row = blockIdx.x;           // 0 .. B*A-1
    const int b   = row >> 11;            // A = 2048

    // cooperative load of rbf strip (64 x 128) into padded LDS; rbf is
    // use-once streaming data -> non-temporal loads (TH=NT)
    const float* rbf_row = rbf + (size_t)row * (NK * NR);
    #pragma unroll
    for (int i = 0; i < 32; ++i) {
        int idx = i * 256 + tid;          // 0..8191
        int r = idx >> 7, c = idx & 127;
        sR[r * PITCH2 + c] = __builtin_nontemporal_load(&rbf_row[idx]);
    }
    if (tid < NK) {
        cd[tid]   = poly_cutoff(D[row * NK + tid]) * NM[row * NK + tid];
        nidx[tid] = N[row * NK + tid];
    }
    __syncthreads();

    const int wave = tid >> 5;
    const int lane = tid & 31;
    const int l16  = lane & 15;
    const int hi   = lane >> 4;

    for (int t = 0; t < 2; ++t) {
        const int nt   = wave * 2 + t;
        const int fcol = nt * 16 + l16;
        const float bias = br[fcol];
        const float ai   = amsij[row * NF + fcol];

        v8f acc[4] = {{}, {}, {}, {}};
        for (int kc = 0; kc < NR / 4; ++kc) {
            const int k0 = kc * 4 + 2 * hi;
            v2f bf;
            bf.x = Wr[(k0    ) * NF + fcol];
            bf.y = Wr[(k0 + 1) * NF + fcol];
            #pragma unroll
            for (int mt = 0; mt < 4; ++mt) {
                v2f af = *(const v2f*)(&sR[(mt * 16 + l16) * PITCH2 + k0]);
                acc[mt] = __builtin_amdgcn_wmma_f32_16x16x4_f32(
                    false, af, false, bf, (short)0, acc[mt], false, false);
            }
        }

        // fused epilogue: bias + cutoff + ai*aj gather + p_out + K-reduction
        float fsum = 0.0f;
        #pragma unroll
        for (int mt = 0; mt < 4; ++mt) {
            #pragma unroll
            for (int v = 0; v < 8; ++v) {
                const int k = mt * 16 + v + 8 * hi;
                float r  = (acc[mt][v] + bias) * cd[k];
                float aj = amsij[(b * NA + nidx[k]) * NF + fcol];  // L2-hot gather
                float m  = r * ai * aj;
                const int poff = (row * NK + k) * NF + fcol;
                // p in/out are use-once 537 MB streams -> non-temporal
                float pin = __builtin_nontemporal_load(&p[poff]);
                __builtin_nontemporal_store(pin + m, &p_out[poff]);
                fsum += m;
            }
        }
        // lanes L and L^16 hold the same f; combine their half-K partial sums
        fsum += __shfl_xor(fsum, 16, 32);
        if (hi == 0)
            a_out[row * NF + fcol] = a[row * NF + fcol] + fsum;
    }
}

// ---------------------------------------------------------------------------
extern "C" void kernel_launch(void* const* d_in, const int* in_sizes, int n_in,
                              void* d_out, int out_size, void* d_ws, size_t ws_size,
                              hipStream_t stream) {
    const float* a     = (const float*)d_in[0];
    const float* p     = (const float*)d_in[1];
    const float* rbf   = (const float*)d_in[2];
    const float* D     = (const float*)d_in[3];
    const int*   N     = (const int*)  d_in[4];
    const float* NM    = (const float*)d_in[5];
    const float* W_rbf = (const float*)d_in[6];
    const float* b_rbf = (const float*)d_in[7];
    const float* W_a1  = (const float*)d_in[8];
    const float* b_a1  = (const float*)d_in[9];
    const float* W_a2  = (const float*)d_in[10];
    const float* b_a2  = (const float*)d_in[11];

    float* a_out = (float*)d_out;                        // B*A*F
    float* p_out = (float*)d_out + (size_t)NB * NA * NF; // B*A*K*F
    float* amsij = (float*)d_ws;                         // B*A*F = 8 MB scratch

    // Kernel 1: a_msij path (two fp32 WMMA GEMMs + softplus), 512 strips
    k_amsij<<<(NB * NA) / 16, 256, 0, stream>>>(a, W_a1, b_a1, W_a2, b_a2, amsij);

    // Kernel 2: fused rbf GEMM + message product + outputs, one block per (b,a)
    k_msg<<<NB * NA, 256, 0, stream>>>(a, p, rbf, D, N, NM, W_rbf, b_rbf,
                                       amsij, a_out, p_out);
}